// IcosphereGradient_38878043963664
// MI455X (gfx1250) — compile-verified
//
#include <hip/hip_runtime.h>
#include <hip/hip_bf16.h>

// Problem constants (reference fixes L=10).
constexpr int B  = 4, F = 5, HS = 1024, S = 2048;
constexpr int BF = B * F;
constexpr size_t PLANE   = (size_t)HS * S;       // x plane per (b,f)
constexpr size_t DSTRIDE = (size_t)F * HS * S;   // out stride between d indices

// Interior tiling: 256 threads (8 wave32), 2 rows per block, 258-wide staged tile.
constexpr int TW    = 256;
constexpr int TH    = 2;
constexpr int LCOLS = TW + 2;        // 258 staged columns (halo)
constexpr int LSTR  = LCOLS + 2;     // 260: pad LDS row stride
constexpr int SROWS = TH + 2;        // 4 staged rows (halo)

#ifndef __has_builtin
#define __has_builtin(x) 0
#endif
#if __has_builtin(__builtin_amdgcn_global_load_async_to_lds_b32)
#define USE_ASYNC_LDS 1
#endif
#if __has_builtin(__builtin_amdgcn_s_wait_asynccnt)
#define HAVE_WAIT_ASYNC_BUILTIN 1
#endif

#ifdef USE_ASYNC_LDS
typedef __attribute__((address_space(1))) int gas_int;   // global
typedef __attribute__((address_space(3))) int las_int;   // LDS
#endif

__device__ __forceinline__ void async_copy_b32(const float* g, float* l) {
#ifdef USE_ASYNC_LDS
  // gfx1250: GLOBAL_LOAD_ASYNC_TO_LDS_B32 (tracked by ASYNCcnt)
  __builtin_amdgcn_global_load_async_to_lds_b32(
      (gas_int*)(float*)g,   // non-const addrspace(1) int*
      (las_int*)l,           // addrspace(3) int*
      /*offset=*/0, /*cpol=*/0);
#else
  *l = *g;  // fallback: global_load_b32 + ds_store_b32
#endif
}

__device__ __forceinline__ void wait_async() {
#ifdef USE_ASYNC_LDS
#ifdef HAVE_WAIT_ASYNC_BUILTIN
  __builtin_amdgcn_s_wait_asynccnt(0);
#else
  asm volatile("s_wait_asynccnt 0x0" ::: "memory");
#endif
#endif
}

// ---------------------------------------------------------------------------
// Interior: j in [1,HS-2], i in [1,S-2].
//   out0 = (x[j-1,i-1]-c)*w0
//   out1 = (x[j+1,i+1]-c)*w1
//   out2 = (x[j-1,i  ]-c)*w2 + (x[j,i+1]-c)*w3
//   out3 = (x[j  ,i-1]-c)*w4 + (x[j+1,i]-c)*w5
// Weights w_inner[k][j-1][i-1] held in registers across the 20 (b,f) planes.
// ---------------------------------------------------------------------------
__global__ __launch_bounds__(TW) void icos_interior_kernel(
    const float* __restrict__ x, const float* __restrict__ w_inner,
    float* __restrict__ out) {
  __shared__ float tile[SROWS * LSTR];
  const int t  = threadIdx.x;
  const int i0 = 1 + (int)blockIdx.x * TW;
  const int j0 = 1 + (int)blockIdx.y * TH;
  const int i  = i0 + t;
  const bool act = (i <= S - 2);

  float w[TH][6];
  if (act) {
#pragma unroll
    for (int r = 0; r < TH; ++r)
#pragma unroll
      for (int k = 0; k < 6; ++k)
        w[r][k] = w_inner[((size_t)k * (HS - 2) + (size_t)(j0 + r - 1)) * (S - 2)
                          + (i - 1)];
  }

  const bool stage_main = (i <= S - 1);
  const int  hrow = t >> 1, hside = t & 1;
  const int  hlc  = hside ? (TW + 1) : 0;   // LDS halo column
  const int  hgi  = i0 - 1 + hlc;           // its global column
  const bool stage_halo = (t < 2 * SROWS) && (hgi <= S - 1);

  for (int bf = 0; bf < BF; ++bf) {
    const int b = bf / F, f = bf % F;
    const float* xp = x + (size_t)bf * PLANE;

    // Stage rows j0-1 .. j0+TH into LDS (async, ASYNCcnt).
    if (stage_main) {
#pragma unroll
      for (int r = 0; r < SROWS; ++r)
        async_copy_b32(xp + (size_t)(j0 - 1 + r) * S + i,
                       &tile[r * LSTR + (t + 1)]);
    }
    if (stage_halo)
      async_copy_b32(xp + (size_t)(j0 - 1 + hrow) * S + hgi,
                     &tile[hrow * LSTR + hlc]);

    // Prefetch next plane (8 MB away -> always cold): global_prefetch_b8.
    if (stage_main && (bf + 1 < BF)) {
      const float* xn = xp + PLANE;
#pragma unroll
      for (int r = 0; r < SROWS; ++r)
        __builtin_prefetch(xn + (size_t)(j0 - 1 + r) * S + i, 0, 1);
    }

    wait_async();
    __syncthreads();

    if (act) {
      const int lc = t + 1;
#pragma unroll
      for (int r = 0; r < TH; ++r) {
        const int   lr    = r + 1;
        const float c     = tile[lr * LSTR + lc];
        const float xm1m1 = tile[(lr - 1) * LSTR + lc - 1];
        const float xm10  = tile[(lr - 1) * LSTR + lc];
        const float x0m1  = tile[lr * LSTR + lc - 1];
        const float x0p1  = tile[lr * LSTR + lc + 1];
        const float xp10  = tile[(lr + 1) * LSTR + lc];
        const float xp1p1 = tile[(lr + 1) * LSTR + lc + 1];
        const int   j     = j0 + r;
        float* o = out + (((size_t)(b * 4) * F + f) * HS + j) * S + i;
        o[0]           = (xm1m1 - c) * w[r][0];
        o[DSTRIDE]     = (xp1p1 - c) * w[r][1];
        o[2 * DSTRIDE] = (xm10 - c) * w[r][2] + (x0p1 - c) * w[r][3];
        o[3 * DSTRIDE] = (x0m1 - c) * w[r][4] + (xp10 - c) * w[r][5];
      }
    }
    __syncthreads();  // protect LDS tile before next plane's staging
  }
}

// ---------------------------------------------------------------------------
// Boundary: rows 0 and HS-1, cols 0 and S-1 (11 regions incl. corners/poles).
// One thread per boundary point; loops over all (b,f) and d.
// ---------------------------------------------------------------------------
__global__ __launch_bounds__(256) void icos_boundary_kernel(
    const float* __restrict__ x, const float* __restrict__ poles,
    const float* __restrict__ w_east, const float* __restrict__ w_cn,
    const float* __restrict__ w_ne, const float* __restrict__ w_nne,
    const float* __restrict__ w_nw, const float* __restrict__ w_nww,
    const float* __restrict__ w_west, const float* __restrict__ w_sw,
    const float* __restrict__ w_sse, const float* __restrict__ w_se,
    const float* __restrict__ w_cs, float* __restrict__ out) {
  const int NB = 2 * S + 2 * (HS - 2);
  const int p  = blockIdx.x * blockDim.x + threadIdx.x;
  if (p >= NB) return;

  int j, i;
  if (p < S)            { j = 0;      i = p; }
  else if (p < 2 * S)   { j = HS - 1; i = p - S; }
  else if (p < 2 * S + (HS - 2)) { j = 1 + (p - 2 * S); i = 0; }
  else                  { j = 1 + (p - 2 * S - (HS - 2)); i = S - 1; }

  for (int b = 0; b < B; ++b) {
    for (int f = 0; f < F; ++f) {
      const int fp = (f + F - 1) % F;   // prev: roll(+1) along features
      const int fn = (f + 1) % F;       // next: roll(-1)
      const float* xb  = x + ((size_t)b * F + f)  * PLANE;
      const float* xbp = x + ((size_t)b * F + fp) * PLANE;
      const float* xbn = x + ((size_t)b * F + fn) * PLANE;
      auto X  = [&](int jj, int ii) { return xb [(size_t)jj * S + ii]; };
      auto Xp = [&](int jj, int ii) { return xbp[(size_t)jj * S + ii]; };
      auto Xn = [&](int jj, int ii) { return xbn[(size_t)jj * S + ii]; };

      const float c = X(j, i);
      float o0, o1, o2, o3;

      if (j == 0) {
        if (i == 0) {  // cs
          o0 = (Xp(HS - 1, HS - 1) - c) * w_cs[0];
          o1 = (X(1, 1) - c) * w_cs[1];
          o2 = (Xp(HS - 1, HS) - c) * w_cs[2] + (X(0, 1) - c) * w_cs[3];
          o3 = (X(1, 0) - c) * w_cs[4];
        } else if (i <= HS - 1) {  // east
          const int e = i - 1; const int W = HS - 1;
          o0 = (Xp(HS - 1, HS + e) - c) * w_east[0 * W + e];
          o1 = (X(1, i + 1) - c) * w_east[1 * W + e];
          o2 = (Xp(HS - 1, HS + 1 + e) - c) * w_east[2 * W + e]
             + (X(0, i + 1) - c) * w_east[3 * W + e];
          o3 = (X(0, i - 1) - c) * w_east[4 * W + e]
             + (X(1, i) - c) * w_east[5 * W + e];
        } else if (i == HS) {  // cn
          o0 = (Xp(HS - 1, S - 1) - c) * w_cn[0];
          o1 = (X(1, HS + 1) - c) * w_cn[1];
          o2 = (X(0, HS + 1) - c) * w_cn[2];
          o3 = (X(0, HS - 1) - c) * w_cn[3] + (X(1, HS) - c) * w_cn[4];
        } else if (i <= S - 2) {  // ne
          const int e = i - HS - 1; const int W = S - HS - 2;
          o0 = (Xp(HS - 2 - e, S - 1) - c) * w_ne[0 * W + e]
             + (Xp(HS - 1 - e, S - 1) - c) * w_ne[1 * W + e];
          o1 = (X(1, i + 1) - c) * w_ne[2 * W + e]
             + (X(1, i) - c) * w_ne[3 * W + e];
          o2 = (X(0, i + 1) - c) * w_ne[4 * W + e];
          o3 = (X(0, i - 1) - c) * w_ne[5 * W + e];
        } else {  // nne (i == S-1)
          o0 = (Xp(0, S - 1) - c) * w_nne[0] + (Xp(1, S - 1) - c) * w_nne[1];
          o1 = (Xn(0, S - 1) - c) * w_nne[2] + (X(1, S - 1) - c) * w_nne[3];
          o2 = (poles[b * 2 + 1] - c) * w_nne[4];
          o3 = (X(0, S - 2) - c) * w_nne[5];
        }
      } else if (j == HS - 1) {
        if (i == 0) {  // sse
          o0 = (Xp(HS - 1, 0) - c) * w_sse[0];
          o1 = (Xn(HS - 1, 0) - c) * w_sse[1];
          o2 = (X(HS - 2, 0) - c) * w_sse[2] + (X(HS - 1, 1) - c) * w_sse[3];
          o3 = (poles[b * 2 + 0] - c) * w_sse[4];
        } else if (i <= HS - 1) {  // sw
          const int e = i - 1; const int W = HS - 1;
          o0 = (X(HS - 2, i - 1) - c) * w_sw[0 * W + e];
          o1 = (Xn(HS - 2 - e, 0) - c) * w_sw[1 * W + e];
          o2 = (X(HS - 2, i) - c) * w_sw[2 * W + e]
             + (X(HS - 1, i + 1) - c) * w_sw[3 * W + e];
          o3 = (X(HS - 1, i - 1) - c) * w_sw[4 * W + e]
             + (Xn(HS - 1 - e, 0) - c) * w_sw[5 * W + e];
        } else if (i <= S - 2) {  // west
          const int e = i - HS; const int W = S - 1 - HS;
          o0 = (X(HS - 2, i - 1) - c) * w_west[0 * W + e];
          o1 = (Xn(0, 1 + e) - c) * w_west[1 * W + e];
          o2 = (X(HS - 2, i) - c) * w_west[2 * W + e]
             + (X(HS - 1, i + 1) - c) * w_west[3 * W + e];
          o3 = (X(HS - 1, i - 1) - c) * w_west[4 * W + e]
             + (Xn(0, e) - c) * w_west[5 * W + e];
        } else {  // nww (i == S-1)
          o0 = (X(HS - 2, S - 2) - c) * w_nww[0];
          o1 = (Xn(0, HS) - c) * w_nww[1];
          o2 = (X(HS - 2, S - 1) - c) * w_nww[2] + (Xn(0, HS + 1) - c) * w_nww[3];
          o3 = (X(HS - 1, S - 2) - c) * w_nww[4] + (Xn(0, HS - 1) - c) * w_nww[5];
        }
      } else if (i == 0) {  // se (j in [1,HS-2])
        const int e = j - 1; const int W = HS - 2;
        o0 = (Xp(HS - 1, HS - 1 - e) - c) * w_se[0 * W + e]
           + (Xp(HS - 1, HS - 2 - e) - c) * w_se[1 * W + e];
        o1 = (X(j, 1) - c) * w_se[2 * W + e] + (X(j + 1, 1) - c) * w_se[3 * W + e];
        o2 = (X(j - 1, 0) - c) * w_se[4 * W + e];
        o3 = (X(j + 1, 0) - c) * w_se[5 * W + e];
      } else {  // nw (i == S-1, j in [1,HS-2])
        const int e = j - 1; const int W = HS - 2;
        o0 = (X(j - 1, S - 2) - c) * w_nw[0 * W + e];
        o1 = (Xn(0, S - 2 - e) - c) * w_nw[1 * W + e];
        o2 = (X(j - 1, S - 1) - c) * w_nw[2 * W + e]
           + (Xn(0, S - 1 - e) - c) * w_nw[3 * W + e];
        o3 = (X(j, S - 2) - c) * w_nw[4 * W + e]
           + (X(j + 1, S - 1) - c) * w_nw[5 * W + e];
      }

      float* o = out + (((size_t)(b * 4) * F + f) * HS + j) * S + i;
      o[0] = o0; o[DSTRIDE] = o1; o[2 * DSTRIDE] = o2; o[3 * DSTRIDE] = o3;
    }
  }
}

extern "C" void kernel_launch(void* const* d_in, const int* in_sizes, int n_in,
                              void* d_out, int out_size, void* d_ws, size_t ws_size,
                              hipStream_t stream) {
  const float* x       = (const float*)d_in[0];
  const float* poles   = (const float*)d_in[1];
  const float* w_inner = (const float*)d_in[2];
  const float* w_east  = (const float*)d_in[3];
  const float* w_cn    = (const float*)d_in[4];
  const float* w_ne    = (const float*)d_in[5];
  const float* w_nne   = (const float*)d_in[6];
  const float* w_nw    = (const float*)d_in[7];
  const float* w_nww   = (const float*)d_in[8];
  const float* w_west  = (const float*)d_in[9];
  const float* w_sw    = (const float*)d_in[10];
  const float* w_sse   = (const float*)d_in[11];
  const float* w_se    = (const float*)d_in[12];
  const float* w_cs    = (const float*)d_in[13];
  float* out = (float*)d_out;

  dim3 gridI((S - 2 + TW - 1) / TW, (HS - 2) / TH);  // 8 x 511 blocks
  icos_interior_kernel<<<gridI, TW, 0, stream>>>(x, w_inner, out);

  const int NB = 2 * S + 2 * (HS - 2);
  icos_boundary_kernel<<<(NB + 255) / 256, 256, 0, stream>>>(
      x, poles, w_east, w_cn, w_ne, w_nne, w_nw, w_nww, w_west, w_sw, w_sse,
      w_se, w_cs, out);
}